// support_sampler_69526930588018
// MI455X (gfx1250) — compile-verified
//
#include <hip/hip_runtime.h>

typedef float v2f __attribute__((ext_vector_type(2)));
typedef float v8f __attribute__((ext_vector_type(8)));

#define OBJ_DIM  1024
#define K_TOTAL  2048
#define NGROUPS  (K_TOTAL / 4)   // 512 K-groups of 4 for 16x16x4 WMMA
#define N_PAIRS  65536

// ---------------------------------------------------------------------------
// Kernel 1: fold the two linear layers.
//   Wc[n][k] = sum_j W2[n][j] * W1[j][k]   (n < 8; cols 8..15 zero-padded)
//   bc[n]    = b2[n] + sum_j W2[n][j] * b1[j]
// Wc is emitted already in V_WMMA_F32_16X16X4_F32 B-operand order:
//   packed[g*64 + lane*2 + c] = Wc[lane&15][4g + 2*(lane>>4) + c]
// so the main kernel's B load is one coalesced float2 per lane.
// ---------------------------------------------------------------------------
__global__ void fold_weights_kernel(const float* __restrict__ W1,
                                    const float* __restrict__ b1,
                                    const float* __restrict__ W2,
                                    const float* __restrict__ b2,
                                    float* __restrict__ WcP,
                                    float* __restrict__ bc) {
  int t = blockIdx.x * blockDim.x + threadIdx.x;
  if (t < NGROUPS * 64) {
    int g   = t >> 6;
    int l   = (t >> 1) & 31;
    int c   = t & 1;
    int col = l & 15;
    int k   = 4 * g + 2 * (l >> 4) + c;
    float acc = 0.0f;
    if (col < 8) {
      const float* w2r = W2 + col * OBJ_DIM;
      for (int j = 0; j < OBJ_DIM; ++j)
        acc = fmaf(w2r[j], W1[(size_t)j * K_TOTAL + k], acc);
    }
    WcP[t] = acc;
  } else if (t < NGROUPS * 64 + 16) {
    int n = t - NGROUPS * 64;
    float acc = 0.0f;
    if (n < 8) {
      const float* w2r = W2 + n * OBJ_DIM;
      for (int j = 0; j < OBJ_DIM; ++j)
        acc = fmaf(w2r[j], b1[j], acc);
      acc += b2[n];
    }
    bc[n] = acc;
  }
}

// ---------------------------------------------------------------------------
// Kernel 2: gather + tall-skinny GEMM with V_WMMA_F32_16X16X4_F32.
// One wave handles 32 pair-rows (two 16x16 D tiles). The A-operand layout
// (lane l holds A[l&15][k0 + 2*(l>>4)], A[l&15][k0 + 2*(l>>4) + 1]) is a
// contiguous float2, so each lane loads its operand fragment DIRECTLY from
// the gathered row — no shuffles, no selects. Loop body per K-group:
//   2x global_load_b64 (A tiles) + 1x global_load_b64 (B) + 2x v_wmma.
// ---------------------------------------------------------------------------
__global__ void __launch_bounds__(256)
pair_gemm_kernel(const float* __restrict__ obj_feats,
                 const long long* __restrict__ pairs,
                 const float* __restrict__ WcP,
                 const float* __restrict__ bc,
                 float* __restrict__ out) {
  const int lane     = threadIdx.x & 31;
  const int wave     = threadIdx.x >> 5;
  const int rowsBase = blockIdx.x * 256 + wave * 32;
  const int half     = lane >> 4;
  const int mrow     = lane & 15;

  // Rows this lane's A fragments come from (one per 16x16 tile).
  const int r0 = rowsBase + mrow;        // tile0: rows base+0..15
  const int r1 = rowsBase + 16 + mrow;   // tile1: rows base+16..31

  const long long s0 = pairs[2 * (long long)r0 + 0];
  const long long o0 = pairs[2 * (long long)r0 + 1];
  const long long s1 = pairs[2 * (long long)r1 + 0];
  const long long o1 = pairs[2 * (long long)r1 + 1];

  // Per-lane K offset inside each 4-wide group: high lane-half takes K+2.
  const int koff = 2 * half;
  const float* a0s = obj_feats + (size_t)s0 * OBJ_DIM + koff;  // K 0..1023
  const float* a1s = obj_feats + (size_t)s1 * OBJ_DIM + koff;
  const float* a0o = obj_feats + (size_t)o0 * OBJ_DIM + koff;  // K 1024..2047
  const float* a1o = obj_feats + (size_t)o1 * OBJ_DIM + koff;

  const v2f* bvec = (const v2f*)WcP + lane;   // + g*32 per K-group

  v8f acc0 = {};  // rows rowsBase + 0..15
  v8f acc1 = {};  // rows rowsBase + 16..31

#pragma unroll 8
  for (int g = 0; g < NGROUPS / 2; ++g) {     // subject half
    v2f a0 = *(const v2f*)(a0s + 4 * g);
    v2f a1 = *(const v2f*)(a1s + 4 * g);
    v2f b  = bvec[g * 32];
    acc0 = __builtin_amdgcn_wmma_f32_16x16x4_f32(false, a0, false, b,
                                                 (short)0, acc0, false, false);
    acc1 = __builtin_amdgcn_wmma_f32_16x16x4_f32(false, a1, false, b,
                                                 (short)0, acc1, false, false);
  }
#pragma unroll 8
  for (int g = 0; g < NGROUPS / 2; ++g) {     // object half
    v2f a0 = *(const v2f*)(a0o + 4 * g);
    v2f a1 = *(const v2f*)(a1o + 4 * g);
    v2f b  = bvec[(g + NGROUPS / 2) * 32];
    acc0 = __builtin_amdgcn_wmma_f32_16x16x4_f32(false, a0, false, b,
                                                 (short)0, acc0, false, false);
    acc1 = __builtin_amdgcn_wmma_f32_16x16x4_f32(false, a1, false, b,
                                                 (short)0, acc1, false, false);
  }

  // D layout: lane = column N, VGPR v = row M (lane-half adds M+=8).
  // Only the first 8 of 16 N columns are real classes.
  const int col = lane & 15;
  if (col < 8) {
    const float bias = bc[col];
#pragma unroll
    for (int v = 0; v < 8; ++v) {
      out[(size_t)(rowsBase + half * 8 + v) * 8 + col]      = acc0[v] + bias;
      out[(size_t)(rowsBase + 16 + half * 8 + v) * 8 + col] = acc1[v] + bias;
    }
  }
}

// ---------------------------------------------------------------------------
extern "C" void kernel_launch(void* const* d_in, const int* in_sizes, int n_in,
                              void* d_out, int out_size, void* d_ws, size_t ws_size,
                              hipStream_t stream) {
  const float*     obj_feats = (const float*)d_in[0];
  const long long* pairs     = (const long long*)d_in[1];
  const float*     W1        = (const float*)d_in[2];
  const float*     b1        = (const float*)d_in[3];
  const float*     W2        = (const float*)d_in[4];
  const float*     b2        = (const float*)d_in[5];
  float*           out       = (float*)d_out;

  float* WcP = (float*)d_ws;              // 512*64 = 32768 floats (128 KB)
  float* bc  = WcP + NGROUPS * 64;        // 16 floats

  int fold_threads = NGROUPS * 64 + 16;
  fold_weights_kernel<<<(fold_threads + 255) / 256, 256, 0, stream>>>(
      W1, b1, W2, b2, WcP, bc);

  pair_gemm_kernel<<<N_PAIRS / 256, 256, 0, stream>>>(
      obj_feats, pairs, WcP, bc, out);
}